// SpatioTemporalBlock_3367254360688
// MI455X (gfx1250) — compile-verified
//
#include <hip/hip_runtime.h>
#include <math.h>

typedef float v2f __attribute__((ext_vector_type(2)));
typedef float v8f __attribute__((ext_vector_type(8)));

#define B_    8
#define CIN   32
#define COUT  32
#define N_    400
#define T_    64
#define FEV   16
#define DNODE 40
#define DK    32

// ---------------------------------------------------------------------------
// Prep: kmat[n][d] = sum_f node_emb[n,f]*Wk[d,f];  acomb = A_static + A_adaptive
// ---------------------------------------------------------------------------
__global__ __launch_bounds__(256) void prep_kernel(
    const float* __restrict__ node_emb, const float* __restrict__ Wk,
    const float* __restrict__ A_static, const float* __restrict__ A_adapt,
    float* __restrict__ kmat, float* __restrict__ acomb)
{
  int i = blockIdx.x * 256 + threadIdx.x;
  if (i < N_ * N_) acomb[i] = A_static[i] + A_adapt[i];
  if (i < N_ * DK) {
    int n = i >> 5, d = i & 31;
    float acc = 0.f;
#pragma unroll
    for (int f = 0; f < DNODE; ++f) acc += node_emb[n * DNODE + f] * Wk[d * DNODE + f];
    kmat[i] = acc;
  }
}

// ---------------------------------------------------------------------------
// Fused spatio-temporal block: one workgroup per (b,t).
// ---------------------------------------------------------------------------
__global__ __launch_bounds__(256) void st_block_kernel(
    const float* __restrict__ traffic,   // (B,32,400,64)
    const float* __restrict__ event_x,   // (B,64,400,16)
    const float* __restrict__ Wf, const float* __restrict__ bf,
    const float* __restrict__ Wg, const float* __restrict__ bg,
    const float* __restrict__ Wq,        // (32,16)
    const float* __restrict__ alpha_p,   // (1,)
    const float* __restrict__ Wr, const float* __restrict__ br,
    const float* __restrict__ Ws, const float* __restrict__ bs,
    const float* __restrict__ kmat,      // (400,32) workspace
    const float* __restrict__ acomb,     // (400,400) workspace
    float* __restrict__ res_out, float* __restrict__ skip_out)
{
  // --- LDS (f32 elements). Odd pitches where WMMA lanes stride across rows.
  __shared__ float lds_kT[DK * N_];          // [d][n]   (B-side: lanes contiguous)
  __shared__ float lds_x [N_ * COUT];        // [n][c]   x_temporal^T (B-side)
  __shared__ float lds_s [16 * 401];         // scores / A_final tile (A-side, pitch 401)
  __shared__ float lds_qt[16 * 33];          // q tile   (A-side, pitch 33)
  __shared__ float lds_o [16 * 33];          // x_gcn tile (A-side, pitch 33)
  __shared__ float lds_p [4][16 * 33];       // GEMM2 deterministic partials
  __shared__ float lds_wf[2048];             // [(i*2+k)*32 + o]
  __shared__ float lds_wg[2048];
  __shared__ float lds_wsT[1024];            // [c*32 + o] (B-side)
  __shared__ float lds_wrT[1024];
  __shared__ float lds_wq[512];              // [d*16 + f]
  __shared__ float lds_tap[8][66];           // [nn][tap*33 + i]
  __shared__ float s_bf[32], s_bg[32], s_br[32], s_bs[32];

  const int tid  = threadIdx.x;
  const int bt   = blockIdx.x;
  const int b    = bt >> 6;                  // /T_
  const int t    = bt & 63;
  const int lane = tid & 31;
  const int wave = tid >> 5;
  const int l16  = lane & 15;
  const int hlf  = lane >> 4;                // 0 or 1

  // ---------------- Phase 0: stage weights & k-matrix ----------------
  for (int j = tid; j < 2048; j += 256) {
    int o = j & 31, ik = j >> 5;             // ik = i*2+k, 0..63
    int i = ik >> 1, k = ik & 1;
    lds_wf[ik * 32 + o] = Wf[(o * 32 + i) * 2 + k];
    lds_wg[ik * 32 + o] = Wg[(o * 32 + i) * 2 + k];
  }
  for (int j = tid; j < 1024; j += 256) {
    int o = j & 31, c = j >> 5;
    lds_wsT[c * 32 + o] = Ws[o * 32 + c];
    lds_wrT[c * 32 + o] = Wr[o * 32 + c];
  }
  for (int j = tid; j < 512; j += 256) lds_wq[j] = Wq[j];
  if (tid < 32) { s_bf[tid] = bf[tid]; s_bg[tid] = bg[tid];
                  s_br[tid] = br[tid]; s_bs[tid] = bs[tid]; }
  for (int j = tid; j < N_ * DK; j += 256) {
    int n = j >> 5, d = j & 31;
    lds_kT[d * N_ + n] = kmat[j];            // transpose to [d][n]
  }
  const float alpha = alpha_p[0];

  // ---------------- Phase 1: gated dilated TCN -> lds_x[n][c] ----------------
  for (int nb = 0; nb < N_; nb += 8) {
    __syncthreads();
    for (int j = tid; j < 512; j += 256) {   // stage 8 nodes x 32 ch x 2 taps
      int i = j & 31, nn = (j >> 5) & 7, tap = j >> 8;
      int tt = t - 2 + 2 * tap;              // tap0 -> t-2 (causal), tap1 -> t
      float v = 0.f;
      if (tt >= 0)
        v = traffic[(((size_t)b * CIN + i) * N_ + (nb + nn)) * T_ + tt];
      lds_tap[nn][tap * 33 + i] = v;
    }
    __syncthreads();
    int c = tid & 31, nn = tid >> 5;
    float f = s_bf[c], g = s_bg[c];
#pragma unroll 8
    for (int i = 0; i < 32; ++i) {
      float x0 = lds_tap[nn][i];
      float x1 = lds_tap[nn][33 + i];
      f += x0 * lds_wf[(2 * i) * 32 + c] + x1 * lds_wf[(2 * i + 1) * 32 + c];
      g += x0 * lds_wg[(2 * i) * 32 + c] + x1 * lds_wg[(2 * i + 1) * 32 + c];
    }
    float sg = 1.f / (1.f + __expf(-g));
    lds_x[(nb + nn) * COUT + c] = tanhf(f) * sg;
  }
  __syncthreads();

  // ---------------- Phase 2: 25 tiles of 16 rows (m) ----------------
  const float inv_sqrt_dk = 0.17677669529663689f;  // 1/sqrt(32)
  for (int mt = 0; mt < 25; ++mt) {
    const int mbase = mt * 16;
    __syncthreads();

    // (a) q tile: q[r][d] = event_x[b,t,m,:] . Wq[d,:]
    for (int j = tid; j < 512; j += 256) {
      int d = j & 31, r = j >> 5;
      const float* ev = event_x + (((size_t)b * T_ + t) * N_ + (mbase + r)) * FEV;
      float acc = 0.f;
#pragma unroll
      for (int f = 0; f < FEV; ++f) acc += ev[f] * lds_wq[d * 16 + f];
      lds_qt[r * 33 + d] = acc;
    }
    __syncthreads();

    // (b) GEMM1: scores[16 x 400] = q(16x32) * k^T(32x400), WMMA f32 16x16x4
    for (int ct = wave; ct < 25; ct += 8) {
      v8f acc = {0.f,0.f,0.f,0.f,0.f,0.f,0.f,0.f};
#pragma unroll
      for (int kk = 0; kk < 8; ++kk) {
        const int d0 = kk * 4 + 2 * hlf;
        v2f a, bm;
        a.x  = lds_qt[l16 * 33 + d0];
        a.y  = lds_qt[l16 * 33 + d0 + 1];
        bm.x = lds_kT[(d0    ) * N_ + ct * 16 + l16];
        bm.y = lds_kT[(d0 + 1) * N_ + ct * 16 + l16];
        acc = __builtin_amdgcn_wmma_f32_16x16x4_f32(false, a, false, bm,
                                                    (short)0, acc, false, false);
      }
#pragma unroll
      for (int v = 0; v < 8; ++v)
        lds_s[(v + 8 * hlf) * 401 + ct * 16 + l16] = acc[v] * inv_sqrt_dk;
    }
    __syncthreads();

    // (c) softmax over 400 cols per row, then A_final = acomb + alpha*softmax
    {
      int r = tid >> 4, ln = tid & 15;       // 16 threads (a half-wave) per row
      float mx = -1e30f;
      for (int j = 0; j < 25; ++j) mx = fmaxf(mx, lds_s[r * 401 + ln + 16 * j]);
      for (int m = 8; m >= 1; m >>= 1) mx = fmaxf(mx, __shfl_xor(mx, m, 32));
      float sum = 0.f;
      for (int j = 0; j < 25; ++j) sum += __expf(lds_s[r * 401 + ln + 16 * j] - mx);
      for (int m = 8; m >= 1; m >>= 1) sum += __shfl_xor(sum, m, 32);
      float inv = 1.f / sum;
      int mrow = mbase + r;
      for (int j = 0; j < 25; ++j) {
        int n = ln + 16 * j;
        float e = __expf(lds_s[r * 401 + n] - mx);
        lds_s[r * 401 + n] = acomb[mrow * N_ + n] + alpha * e * inv;
      }
    }
    __syncthreads();

    // (d) GEMM2: out(16x32) = A_final(16x400) * x^T(400x32); 2 ctiles x 4 kparts
    {
      const int ctile = wave & 1, kpart = wave >> 1;     // kpart 0..3
      v8f acc = {0.f,0.f,0.f,0.f,0.f,0.f,0.f,0.f};
      for (int kk = kpart * 25; kk < kpart * 25 + 25; ++kk) {
        const int n0 = kk * 4 + 2 * hlf;
        v2f a, bm;
        a.x  = lds_s[l16 * 401 + n0];
        a.y  = lds_s[l16 * 401 + n0 + 1];
        bm.x = lds_x[(n0    ) * COUT + ctile * 16 + l16];
        bm.y = lds_x[(n0 + 1) * COUT + ctile * 16 + l16];
        acc = __builtin_amdgcn_wmma_f32_16x16x4_f32(false, a, false, bm,
                                                    (short)0, acc, false, false);
      }
#pragma unroll
      for (int v = 0; v < 8; ++v)
        lds_p[kpart][(v + 8 * hlf) * 33 + ctile * 16 + l16] = acc[v];
    }
    __syncthreads();
    for (int j = tid; j < 512; j += 256) {   // deterministic k-partial reduce
      int m = j >> 5, c = j & 31;
      int idx = m * 33 + c;
      lds_o[idx] = lds_p[0][idx] + lds_p[1][idx] + lds_p[2][idx] + lds_p[3][idx];
    }
    __syncthreads();

    // (e) GEMM3: skip = out*Ws^T + bs ; res = out*Wr^T + br + traffic
    if (wave < 4) {
      const int isres = wave >> 1, otile = wave & 1;
      const float* W = isres ? lds_wrT : lds_wsT;
      v8f acc = {0.f,0.f,0.f,0.f,0.f,0.f,0.f,0.f};
#pragma unroll
      for (int kk = 0; kk < 8; ++kk) {
        const int c0 = kk * 4 + 2 * hlf;
        v2f a, bm;
        a.x  = lds_o[l16 * 33 + c0];
        a.y  = lds_o[l16 * 33 + c0 + 1];
        bm.x = W[(c0    ) * 32 + otile * 16 + l16];
        bm.y = W[(c0 + 1) * 32 + otile * 16 + l16];
        acc = __builtin_amdgcn_wmma_f32_16x16x4_f32(false, a, false, bm,
                                                    (short)0, acc, false, false);
      }
#pragma unroll
      for (int v = 0; v < 8; ++v) {
        int m = mbase + v + 8 * hlf;
        int o = otile * 16 + l16;
        size_t gidx = (((size_t)b * 32 + o) * N_ + m) * T_ + t;
        if (isres) res_out[gidx]  = acc[v] + s_br[o] + traffic[gidx];
        else       skip_out[gidx] = acc[v] + s_bs[o];
      }
    }
  }
}

// ---------------------------------------------------------------------------
extern "C" void kernel_launch(void* const* d_in, const int* in_sizes, int n_in,
                              void* d_out, int out_size, void* d_ws, size_t ws_size,
                              hipStream_t stream)
{
  (void)in_sizes; (void)n_in; (void)out_size; (void)ws_size;
  const float* traffic  = (const float*)d_in[0];
  const float* event_x  = (const float*)d_in[1];
  const float* node_emb = (const float*)d_in[2];
  const float* A_static = (const float*)d_in[3];
  const float* A_adapt  = (const float*)d_in[4];
  const float* Wf       = (const float*)d_in[5];
  const float* bf       = (const float*)d_in[6];
  const float* Wg       = (const float*)d_in[7];
  const float* bg       = (const float*)d_in[8];
  const float* Wq       = (const float*)d_in[9];
  const float* Wk       = (const float*)d_in[10];
  const float* alpha    = (const float*)d_in[11];
  const float* Wr       = (const float*)d_in[12];
  const float* br       = (const float*)d_in[13];
  const float* Ws       = (const float*)d_in[14];
  const float* bs       = (const float*)d_in[15];

  float* kmat  = (float*)d_ws;                    // 400*32 floats
  float* acomb = kmat + N_ * DK;                  // 400*400 floats
  float* res   = (float*)d_out;                   // (8,32,400,64)
  float* skip  = res + (size_t)B_ * CIN * N_ * T_;

  prep_kernel<<<(N_ * N_ + 255) / 256, 256, 0, stream>>>(
      node_emb, Wk, A_static, A_adapt, kmat, acomb);
  st_block_kernel<<<B_ * T_, 256, 0, stream>>>(
      traffic, event_x, Wf, bf, Wg, bg, Wq, alpha, Wr, br, Ws, bs,
      kmat, acomb, res, skip);
}